// DeeperGCN_63196148793950
// MI455X (gfx1250) — compile-verified
//
#include <hip/hip_runtime.h>

#define N_NODES_C 50000
#define N_EDGES_C 500000
#define HID 128
#define EDGE_C 64
#define IN_C 128
#define TASKS 8
#define LAYERS 3

typedef float v2f __attribute__((ext_vector_type(2)));
typedef float v8f __attribute__((ext_vector_type(8)));

enum { A_PLAIN = 0, A_ADD2 = 1, A_BNRELU = 2 };
enum { E_STORE = 0, E_EDGE = 1 };

// Generic fp32 WMMA GEMM: C[M x Ntot] = op_A(A)[M x K] * B[K x Ntot] + bias
// Block: 256 threads = 8 waves; block tile 64(M) x 128(N); wave tile 16 x 64.
// K staged through LDS in 32-wide slabs.
__global__ __launch_bounds__(256)
void gemm_wmma_f32(const float* __restrict__ A,
                   const float* __restrict__ A2,     // A_ADD2: second addend
                   const float* __restrict__ scale,  // A_BNRELU: per-K scale
                   const float* __restrict__ shift,  // A_BNRELU: per-K shift
                   const float* __restrict__ B,
                   const float* __restrict__ bias,
                   float* __restrict__ C,
                   int M, int K, int Ntot,
                   int a_mode, int e_mode,
                   const int* __restrict__ srcIdx,
                   const int* __restrict__ dstIdx,
                   const float* __restrict__ hgather,
                   unsigned* __restrict__ nodemax)
{
    __shared__ float As[64][36];   // 64x32 slab, pad 36 (conflict-free frag reads)
    __shared__ float Bs[32][136];  // 32x128 slab, pad 136

    const int tid  = threadIdx.x;
    const int lane = tid & 31;
    const int wid  = tid >> 5;
    const int wrow = wid & 3;            // 4 wave-rows of 16
    const int wcol = wid >> 2;           // 2 wave-cols of 64
    const int lm   = lane & 15;          // M (A) / N (B,C) index within tile
    const int lk   = (lane >> 4) << 1;   // K sub-offset: 0 or 2

    const int block_m = blockIdx.x * 64;
    const int block_n = blockIdx.y * 128;

    v8f acc[4] = {};

    for (int k0 = 0; k0 < K; k0 += 32) {
        // ---- stage A (64 x 32) with fused A-mode transform ----
        #pragma unroll
        for (int i = 0; i < 2; ++i) {
            int f4  = tid + i * 256;     // 0..511 float4 slots
            int row = f4 >> 3;
            int col = (f4 & 7) << 2;
            int gm  = block_m + row;
            float v0 = 0.f, v1 = 0.f, v2 = 0.f, v3 = 0.f;
            if (gm < M) {
                float4 t4 = *(const float4*)(A + (size_t)gm * K + (k0 + col));
                v0 = t4.x; v1 = t4.y; v2 = t4.z; v3 = t4.w;
                if (a_mode == A_ADD2) {
                    float4 u4 = *(const float4*)(A2 + (size_t)gm * K + (k0 + col));
                    v0 += u4.x; v1 += u4.y; v2 += u4.z; v3 += u4.w;
                } else if (a_mode == A_BNRELU) {
                    float4 sc = *(const float4*)(scale + k0 + col);
                    float4 sh = *(const float4*)(shift + k0 + col);
                    v0 = fmaxf(v0 * sc.x + sh.x, 0.f);
                    v1 = fmaxf(v1 * sc.y + sh.y, 0.f);
                    v2 = fmaxf(v2 * sc.z + sh.z, 0.f);
                    v3 = fmaxf(v3 * sc.w + sh.w, 0.f);
                }
            }
            As[row][col]     = v0;
            As[row][col + 1] = v1;
            As[row][col + 2] = v2;
            As[row][col + 3] = v3;
        }
        // ---- stage B (32 x 128) ----
        #pragma unroll
        for (int i = 0; i < 4; ++i) {
            int f4  = tid + i * 256;     // 0..1023 float4 slots
            int row = f4 >> 5;
            int col = (f4 & 31) << 2;
            float4 t4 = *(const float4*)(B + (size_t)(k0 + row) * Ntot + block_n + col);
            Bs[row][col]     = t4.x;
            Bs[row][col + 1] = t4.y;
            Bs[row][col + 2] = t4.z;
            Bs[row][col + 3] = t4.w;
        }
        __syncthreads();

        // ---- 16x16x4 f32 WMMA over the slab ----
        #pragma unroll
        for (int kk = 0; kk < 32; kk += 4) {
            // A frag: lane holds A[M=lm][K=kk+lk .. kk+lk+1]
            v2f a = *(const v2f*)&As[wrow * 16 + lm][kk + lk];
            #pragma unroll
            for (int t = 0; t < 4; ++t) {
                int n = wcol * 64 + t * 16 + lm;
                v2f b;  // B frag: lane holds B[K=kk+lk+v][N=lm]
                b.x = Bs[kk + lk][n];
                b.y = Bs[kk + lk + 1][n];
                acc[t] = __builtin_amdgcn_wmma_f32_16x16x4_f32(
                    false, a, false, b, (short)0, acc[t], false, false);
            }
        }
        __syncthreads();
    }

    // ---- epilogue (C/D layout: VGPR v -> M = v + 8*(lane>=16), N = lane&15) ----
    const int tile_m = block_m + wrow * 16 + ((lane >> 4) << 3);
    #pragma unroll
    for (int t = 0; t < 4; ++t) {
        int n = block_n + wcol * 64 + t * 16 + lm;
        float bv = bias ? bias[n] : 0.0f;
        #pragma unroll
        for (int v = 0; v < 8; ++v) {
            int m = tile_m + v;
            if (m < M) {
                float val = acc[t][v] + bv;
                if (e_mode == E_EDGE) {
                    // msg = relu(h[src] + ea) + eps ; seg-max over dst
                    int s = srcIdx[m];
                    int d = dstIdx[m];
                    float mg = fmaxf(val + hgather[(size_t)s * HID + n], 0.0f) + 1e-7f;
                    C[(size_t)m * Ntot + n] = mg;
                    // mg > 0 so float bits are monotonic as unsigned
                    atomicMax(&nodemax[(size_t)d * HID + n], __float_as_uint(mg));
                } else {
                    C[(size_t)m * Ntot + n] = val;
                }
            }
        }
    }
}

// ex = exp(msg - max[dst]) ; denom[dst] += ex
__global__ __launch_bounds__(256)
void edge_exp_denom(const float* __restrict__ msg, const int* __restrict__ dst,
                    const float* __restrict__ nodemax, float* __restrict__ denom, int E)
{
    int gid = blockIdx.x * blockDim.x + threadIdx.x;
    int e = gid >> 5;
    if (e >= E) return;
    int c = (gid & 31) << 2;
    int d = dst[e];
    float4 m4 = *(const float4*)(msg + (size_t)e * HID + c);
    float4 x4 = *(const float4*)(nodemax + (size_t)d * HID + c);
    float* dn = denom + (size_t)d * HID + c;
    atomicAdd(dn + 0, expf(m4.x - x4.x));
    atomicAdd(dn + 1, expf(m4.y - x4.y));
    atomicAdd(dn + 2, expf(m4.z - x4.z));
    atomicAdd(dn + 3, expf(m4.w - x4.w));
}

// agg[dst] += msg * exp(msg - max[dst]) / (denom[dst] + 1e-16)
__global__ __launch_bounds__(256)
void edge_aggregate(const float* __restrict__ msg, const int* __restrict__ dst,
                    const float* __restrict__ nodemax, const float* __restrict__ denom,
                    float* __restrict__ agg, int E)
{
    int gid = blockIdx.x * blockDim.x + threadIdx.x;
    int e = gid >> 5;
    if (e >= E) return;
    int c = (gid & 31) << 2;
    int d = dst[e];
    float4 m4 = *(const float4*)(msg + (size_t)e * HID + c);
    float4 x4 = *(const float4*)(nodemax + (size_t)d * HID + c);
    float4 dn = *(const float4*)(denom + (size_t)d * HID + c);
    float* ag = agg + (size_t)d * HID + c;
    atomicAdd(ag + 0, m4.x * (expf(m4.x - x4.x) / (dn.x + 1e-16f)));
    atomicAdd(ag + 1, m4.y * (expf(m4.y - x4.y) / (dn.y + 1e-16f)));
    atomicAdd(ag + 2, m4.z * (expf(m4.z - x4.z) / (dn.z + 1e-16f)));
    atomicAdd(ag + 3, m4.w * (expf(m4.w - x4.w) / (dn.w + 1e-16f)));
}

// per-column sum / sumsq (blockDim.x must equal C)
__global__
void bn_stats(const float* __restrict__ X, float* __restrict__ sum,
              float* __restrict__ sumsq, int N, int C)
{
    int c = threadIdx.x;
    float s = 0.f, q = 0.f;
    for (int r = blockIdx.x; r < N; r += gridDim.x) {
        float v = X[(size_t)r * C + c];
        s += v; q += v * v;
    }
    atomicAdd(&sum[c], s);
    atomicAdd(&sumsq[c], q);
}

__global__
void bn_finalize(const float* __restrict__ sum, const float* __restrict__ sumsq,
                 const float* __restrict__ g, const float* __restrict__ b,
                 float* __restrict__ scale, float* __restrict__ shift, int N, int C)
{
    int c = threadIdx.x + blockIdx.x * blockDim.x;
    if (c < C) {
        float inv_n = 1.0f / (float)N;
        float mean = sum[c] * inv_n;
        float var  = sumsq[c] * inv_n - mean * mean;
        float sc   = g[c] * rsqrtf(var + 1e-5f);
        scale[c] = sc;
        shift[c] = b[c] - mean * sc;
    }
}

// h_out = relu(y*scale+shift) (+ h_prev)
__global__ __launch_bounds__(256)
void bn_relu_residual(const float* __restrict__ y, const float* __restrict__ scale,
                      const float* __restrict__ shift, const float* __restrict__ hprev,
                      float* __restrict__ hout, int total)
{
    int i = blockIdx.x * blockDim.x + threadIdx.x;
    if (i >= total) return;
    int c = i & (HID - 1);
    float r = fmaxf(y[i] * scale[c] + shift[c], 0.0f);
    hout[i] = hprev ? r + hprev[i] : r;
}

// out[n,t] = b[t] + sum_{l,c} h_l[n,c] * W[l*HID+c, t]
__global__ __launch_bounds__(256)
void pred_head(const float* __restrict__ h0, const float* __restrict__ h1,
               const float* __restrict__ h2, const float* __restrict__ W,
               const float* __restrict__ b, float* __restrict__ out, int N)
{
    __shared__ float Ws[LAYERS * HID * TASKS];  // 12 KB
    for (int i = threadIdx.x; i < LAYERS * HID * TASKS; i += 256) Ws[i] = W[i];
    __syncthreads();
    int n = blockIdx.x * 256 + threadIdx.x;
    if (n >= N) return;
    float acc[TASKS];
    #pragma unroll
    for (int t = 0; t < TASKS; ++t) acc[t] = b[t];
    const float* hs[3] = { h0, h1, h2 };
    for (int l = 0; l < LAYERS; ++l) {
        const float* hr = hs[l] + (size_t)n * HID;
        for (int c = 0; c < HID; ++c) {
            float xv = hr[c];
            const float* w = &Ws[(l * HID + c) * TASKS];
            #pragma unroll
            for (int t = 0; t < TASKS; ++t) acc[t] += xv * w[t];
        }
    }
    #pragma unroll
    for (int t = 0; t < TASKS; ++t) out[(size_t)n * TASKS + t] = acc[t];
}

extern "C" void kernel_launch(void* const* d_in, const int* in_sizes, int n_in,
                              void* d_out, int out_size, void* d_ws, size_t ws_size,
                              hipStream_t stream)
{
    const float* x     = (const float*)d_in[0];
    const int*   eidx  = (const int*)d_in[1];
    const float* eattr = (const float*)d_in[2];
    const float* encW  = (const float*)d_in[3];
    const float* encB  = (const float*)d_in[4];
    const float* leW   = (const float*)d_in[5];
    const float* leB   = (const float*)d_in[6];
    const float* W1    = (const float*)d_in[7];
    const float* b1    = (const float*)d_in[8];
    const float* ng    = (const float*)d_in[9];
    const float* nb    = (const float*)d_in[10];
    const float* W2    = (const float*)d_in[11];
    const float* b2    = (const float*)d_in[12];
    const float* normg = (const float*)d_in[13];
    const float* normb = (const float*)d_in[14];
    const float* predW = (const float*)d_in[15];
    const float* predB = (const float*)d_in[16];
    float* out = (float*)d_out;

    const int N = N_NODES_C, E = N_EDGES_C;
    const int* src = eidx;
    const int* dst = eidx + E;

    float* ws = (float*)d_ws;
    size_t o = 0;
    float* henc = ws + o; o += (size_t)N * HID;
    float* hb[3];
    for (int l = 0; l < 3; ++l) { hb[l] = ws + o; o += (size_t)N * HID; }
    float* aggp  = ws + o; o += (size_t)N * HID;
    float* denom = ws + o; o += (size_t)N * HID;
    unsigned* nodemax = (unsigned*)(ws + o); o += (size_t)N * HID;
    float* zb = ws + o; o += (size_t)N * 2 * HID;
    float* yb = ws + o; o += (size_t)N * HID;
    float* bnsum = ws + o; o += 256;
    float* bnsq  = ws + o; o += 256;
    float* scl   = ws + o; o += 256;
    float* sft   = ws + o; o += 256;
    float* msg   = ws + o; o += (size_t)E * HID;

    dim3 blk(256);

    // encoder: h = x @ enc_W + enc_b
    gemm_wmma_f32<<<dim3((N + 63) / 64, 1), blk, 0, stream>>>(
        x, nullptr, nullptr, nullptr, encW, encB, henc,
        N, IN_C, HID, A_PLAIN, E_STORE, nullptr, nullptr, nullptr, nullptr);

    const float* hprev = henc;
    for (int l = 0; l < LAYERS; ++l) {
        hipMemsetAsync(nodemax, 0, (size_t)N * HID * 4, stream);
        hipMemsetAsync(denom,   0, (size_t)N * HID * 4, stream);
        hipMemsetAsync(aggp,    0, (size_t)N * HID * 4, stream);

        // edge GEMM fused with msg construction + segment-max
        gemm_wmma_f32<<<dim3((E + 63) / 64, 1), blk, 0, stream>>>(
            eattr, nullptr, nullptr, nullptr,
            leW + (size_t)l * EDGE_C * HID, leB + (size_t)l * HID, msg,
            E, EDGE_C, HID, A_PLAIN, E_EDGE, src, dst, hprev, nodemax);

        int tot = E * 32;  // one thread per 4 features per edge
        edge_exp_denom<<<(tot + 255) / 256, blk, 0, stream>>>(
            msg, dst, (const float*)nodemax, denom, E);
        edge_aggregate<<<(tot + 255) / 256, blk, 0, stream>>>(
            msg, dst, (const float*)nodemax, denom, aggp, E);

        // z = (h + agg) @ W1 + b1   [N x 256]
        gemm_wmma_f32<<<dim3((N + 63) / 64, 2), blk, 0, stream>>>(
            hprev, aggp, nullptr, nullptr,
            W1 + (size_t)l * HID * 2 * HID, b1 + (size_t)l * 2 * HID, zb,
            N, HID, 2 * HID, A_ADD2, E_STORE, nullptr, nullptr, nullptr, nullptr);

        // inner BN (256 cols)
        hipMemsetAsync(bnsum, 0, 256 * 4, stream);
        hipMemsetAsync(bnsq,  0, 256 * 4, stream);
        bn_stats<<<512, 256, 0, stream>>>(zb, bnsum, bnsq, N, 2 * HID);
        bn_finalize<<<1, 256, 0, stream>>>(bnsum, bnsq,
            ng + (size_t)l * 2 * HID, nb + (size_t)l * 2 * HID, scl, sft, N, 2 * HID);

        // y = relu(bn(z)) @ W2 + b2   (BN+ReLU fused into A-load)
        gemm_wmma_f32<<<dim3((N + 63) / 64, 1), blk, 0, stream>>>(
            zb, nullptr, scl, sft,
            W2 + (size_t)l * 2 * HID * HID, b2 + (size_t)l * HID, yb,
            N, 2 * HID, HID, A_BNRELU, E_STORE, nullptr, nullptr, nullptr, nullptr);

        // outer BN (128 cols) + ReLU + residual
        hipMemsetAsync(bnsum, 0, 256 * 4, stream);
        hipMemsetAsync(bnsq,  0, 256 * 4, stream);
        bn_stats<<<512, 128, 0, stream>>>(yb, bnsum, bnsq, N, HID);
        bn_finalize<<<1, 128, 0, stream>>>(bnsum, bnsq,
            normg + (size_t)l * HID, normb + (size_t)l * HID, scl, sft, N, HID);
        bn_relu_residual<<<((N * HID) + 255) / 256, blk, 0, stream>>>(
            yb, scl, sft, l == 0 ? nullptr : hprev, hb[l], N * HID);

        hprev = hb[l];
    }

    pred_head<<<(N + 255) / 256, blk, 0, stream>>>(
        hb[0], hb[1], hb[2], predW, predB, out, N);
}